// MultiSubjectMultiTexture_44349832299088
// MI455X (gfx1250) — compile-verified
//
#include <hip/hip_runtime.h>
#include <math.h>

#define AS1 __attribute__((address_space(1)))
#define AS3 __attribute__((address_space(3)))

#if defined(__has_builtin)
# if __has_builtin(__builtin_amdgcn_global_load_async_to_lds_b32)
#  define HAVE_ASYNC_LDS 1
# endif
#endif

static constexpr int    RN       = 2097152;
static constexpr int    CCH      = 16;
static constexpr int    MM       = 8;
static constexpr size_t T0_ELEMS = (size_t)MM * CCH * 512 * 512; // 33,554,432
static constexpr size_t T1_ELEMS = (size_t)MM * CCH * 256 * 256; //  8,388,608

// ---------------------------------------------------------------------------
// Pass 1: channel transpose [M,C,H,W] -> [M,H,W,C].
// Templated on the (square) texture edge so all index math is shifts/masks.
// One wave handles a 32-wide x-strip of one (m,y) row; the 16 channel planes
// are pulled straight into LDS with async-to-LDS DMA ops (CDNA5 path), then
// written back as contiguous 64B per texel.
// ---------------------------------------------------------------------------
template <int Wd>
__global__ void transpose_mchw_mhwc(const float* __restrict__ src,
                                    float* __restrict__ dst) {
  constexpr int HW           = Wd * Wd;
  constexpr int stripsPerRow = Wd >> 5;          // power of two
  constexpr int totalStrips  = MM * Wd * stripsPerRow;

  const int lane      = threadIdx.x & 31;
  const int waveInBlk = threadIdx.x >> 5;
  const int wave      = blockIdx.x * (blockDim.x >> 5) + waveInBlk;
  if (wave >= totalStrips) return;

  const int xb = wave & (stripsPerRow - 1);
  const int t  = wave / stripsPerRow;            // shift (pow2)
  const int y  = t & (Wd - 1);
  const int m  = t / Wd;                         // shift (pow2)
  const int x  = (xb << 5) + lane;

  const float* gsrc = src + (size_t)m * CCH * HW + (size_t)y * Wd + x;

  // stride 17 floats per lane -> conflict-free LDS readback (17 coprime w/ 64)
  __shared__ float lds[8 * 32 * 17];
  float* myLds = &lds[waveInBlk * (32 * 17) + lane * 17];

#if HAVE_ASYNC_LDS
  #pragma unroll
  for (int c = 0; c < CCH; ++c) {
    __builtin_amdgcn_global_load_async_to_lds_b32(
        (AS1 int*)(gsrc + (size_t)c * HW),
        (AS3 int*)(&myLds[c]),
        /*offset=*/0, /*cpol=*/0);
  }
# if __has_builtin(__builtin_amdgcn_s_wait_asynccnt)
  __builtin_amdgcn_s_wait_asynccnt(0);
# else
  asm volatile("s_wait_asynccnt 0x0" ::: "memory");
# endif
#else
  #pragma unroll
  for (int c = 0; c < CCH; ++c)
    myLds[c] = __builtin_nontemporal_load(gsrc + (size_t)c * HW);
#endif

  float4 o[4];
  #pragma unroll
  for (int j = 0; j < 4; ++j) {
    o[j].x = myLds[4 * j + 0];
    o[j].y = myLds[4 * j + 1];
    o[j].z = myLds[4 * j + 2];
    o[j].w = myLds[4 * j + 3];
  }
  // keep transposed texture resident (regular-temporal stores -> L2)
  float4* dp = (float4*)(dst + ((size_t)m * HW + (size_t)y * Wd + x) * CCH);
  #pragma unroll
  for (int j = 0; j < 4; ++j) dp[j] = o[j];
}

// ---------------------------------------------------------------------------
// Trilinear helpers
// ---------------------------------------------------------------------------
__device__ __forceinline__ float4 lerp4(const float4 a, const float4 b, const float w) {
  const float omw = 1.0f - w;
  float4 r;
  r.x = a.x * omw + b.x * w;
  r.y = a.y * omw + b.y * w;
  r.z = a.z * omw + b.z * w;
  r.w = a.w * omw + b.w * w;
  return r;
}

// ---------------------------------------------------------------------------
// Pass 2: trilinear sample from the channel-transposed textures.
// Each corner fetch = 64B contiguous (4 x global_load_b128).
// ---------------------------------------------------------------------------
__global__ void sample_t(const float* __restrict__ uv, const int* __restrict__ idcs,
                         const float* __restrict__ sid, const float* __restrict__ t0,
                         const float* __restrict__ t1, float* __restrict__ out) {
  const int r = blockIdx.x * blockDim.x + threadIdx.x;
  if (r >= RN) return;

  const float u = __builtin_nontemporal_load(uv + 2 * (size_t)r + 0);
  const float v = __builtin_nontemporal_load(uv + 2 * (size_t)r + 1);
  const float z = __builtin_nontemporal_load(sid + r);
  const bool is0 = (__builtin_nontemporal_load(idcs + r) == 0);

  const float* base = is0 ? t0 : t1;
  const int W       = is0 ? 512 : 256;   // H == W
  const float wm1   = (float)(W - 1);

  float fx = fminf(fmaxf((u + 1.0f) * 0.5f * wm1, 0.0f), wm1);
  float fy = fminf(fmaxf((v + 1.0f) * 0.5f * wm1, 0.0f), wm1);
  float fd = fminf(fmaxf((z + 1.0f) * 0.5f * 7.0f, 0.0f), 7.0f);

  const int x0 = (int)floorf(fx); const int x1 = min(x0 + 1, W - 1);
  const int y0 = (int)floorf(fy); const int y1 = min(y0 + 1, W - 1);
  const int d0 = (int)floorf(fd); const int d1 = min(d0 + 1, MM - 1);
  const float wx = fx - (float)x0;
  const float wy = fy - (float)y0;
  const float wd = fd - (float)d0;

  auto cp = [&](int dd, int yy, int xx) {
    return (const float4*)(base + (((size_t)dd * W + yy) * W + xx) * CCH);
  };
  const float4* p000 = cp(d0, y0, x0); const float4* p001 = cp(d0, y0, x1);
  const float4* p010 = cp(d0, y1, x0); const float4* p011 = cp(d0, y1, x1);
  const float4* p100 = cp(d1, y0, x0); const float4* p101 = cp(d1, y0, x1);
  const float4* p110 = cp(d1, y1, x0); const float4* p111 = cp(d1, y1, x1);

  #pragma unroll
  for (int j = 0; j < 4; ++j) {
    float4 v00 = lerp4(p000[j], p001[j], wx);
    float4 v01 = lerp4(p010[j], p011[j], wx);
    float4 v10 = lerp4(p100[j], p101[j], wx);
    float4 v11 = lerp4(p110[j], p111[j], wx);
    float4 v0  = lerp4(v00, v01, wy);
    float4 v1  = lerp4(v10, v11, wy);
    float4 res = lerp4(v0, v1, wd);
    const size_t c = 4 * (size_t)j;
    __builtin_nontemporal_store(res.x, out + (c + 0) * (size_t)RN + r);
    __builtin_nontemporal_store(res.y, out + (c + 1) * (size_t)RN + r);
    __builtin_nontemporal_store(res.z, out + (c + 2) * (size_t)RN + r);
    __builtin_nontemporal_store(res.w, out + (c + 3) * (size_t)RN + r);
  }
}

// ---------------------------------------------------------------------------
// Fallback: direct gather from [M,C,H,W] layout (used if workspace too small)
// ---------------------------------------------------------------------------
__global__ void sample_direct(const float* __restrict__ uv, const int* __restrict__ idcs,
                              const float* __restrict__ sid, const float* __restrict__ t0,
                              const float* __restrict__ t1, float* __restrict__ out) {
  const int r = blockIdx.x * blockDim.x + threadIdx.x;
  if (r >= RN) return;

  const float u = uv[2 * (size_t)r + 0];
  const float v = uv[2 * (size_t)r + 1];
  const float z = sid[r];
  const bool is0 = (idcs[r] == 0);

  const float* base = is0 ? t0 : t1;
  const int W  = is0 ? 512 : 256;
  const int HW = W * W;
  const float wm1 = (float)(W - 1);

  float fx = fminf(fmaxf((u + 1.0f) * 0.5f * wm1, 0.0f), wm1);
  float fy = fminf(fmaxf((v + 1.0f) * 0.5f * wm1, 0.0f), wm1);
  float fd = fminf(fmaxf((z + 1.0f) * 0.5f * 7.0f, 0.0f), 7.0f);

  const int x0 = (int)floorf(fx); const int x1 = min(x0 + 1, W - 1);
  const int y0 = (int)floorf(fy); const int y1 = min(y0 + 1, W - 1);
  const int d0 = (int)floorf(fd); const int d1 = min(d0 + 1, MM - 1);
  const float wx = fx - (float)x0;
  const float wy = fy - (float)y0;
  const float wd = fd - (float)d0;

  auto off = [&](int dd, int yy, int xx) {
    return (size_t)dd * CCH * HW + (size_t)yy * W + xx;
  };
  const size_t o000 = off(d0, y0, x0), o001 = off(d0, y0, x1);
  const size_t o010 = off(d0, y1, x0), o011 = off(d0, y1, x1);
  const size_t o100 = off(d1, y0, x0), o101 = off(d1, y0, x1);
  const size_t o110 = off(d1, y1, x0), o111 = off(d1, y1, x1);

  #pragma unroll
  for (int c = 0; c < CCH; ++c) {
    const size_t co = (size_t)c * HW;
    const float g000 = base[o000 + co], g001 = base[o001 + co];
    const float g010 = base[o010 + co], g011 = base[o011 + co];
    const float g100 = base[o100 + co], g101 = base[o101 + co];
    const float g110 = base[o110 + co], g111 = base[o111 + co];
    const float v00 = g000 * (1.0f - wx) + g001 * wx;
    const float v01 = g010 * (1.0f - wx) + g011 * wx;
    const float v10 = g100 * (1.0f - wx) + g101 * wx;
    const float v11 = g110 * (1.0f - wx) + g111 * wx;
    const float v0  = v00 * (1.0f - wy) + v01 * wy;
    const float v1  = v10 * (1.0f - wy) + v11 * wy;
    __builtin_nontemporal_store(v0 * (1.0f - wd) + v1 * wd,
                                out + (size_t)c * RN + r);
  }
}

// ---------------------------------------------------------------------------
extern "C" void kernel_launch(void* const* d_in, const int* in_sizes, int n_in,
                              void* d_out, int out_size, void* d_ws, size_t ws_size,
                              hipStream_t stream) {
  (void)in_sizes; (void)n_in; (void)out_size;
  const float* uv   = (const float*)d_in[0];
  const int*   idcs = (const int*)d_in[1];
  const float* sid  = (const float*)d_in[2];
  const float* tex0 = (const float*)d_in[3];
  const float* tex1 = (const float*)d_in[4];
  float* out = (float*)d_out;

  const size_t wsNeed = (T0_ELEMS + T1_ELEMS) * sizeof(float);
  const int threads = 256;
  const int sampleBlocks = (RN + threads - 1) / threads;

  if (ws_size >= wsNeed) {
    float* t0t = (float*)d_ws;
    float* t1t = t0t + T0_ELEMS;

    const int strips0 = MM * 512 * (512 / 32);       // 65536 waves
    const int blocks0 = (strips0 + 7) / 8;           // 8 waves / block
    transpose_mchw_mhwc<512><<<blocks0, 256, 0, stream>>>(tex0, t0t);

    const int strips1 = MM * 256 * (256 / 32);       // 16384 waves
    const int blocks1 = (strips1 + 7) / 8;
    transpose_mchw_mhwc<256><<<blocks1, 256, 0, stream>>>(tex1, t1t);

    sample_t<<<sampleBlocks, threads, 0, stream>>>(uv, idcs, sid, t0t, t1t, out);
  } else {
    sample_direct<<<sampleBlocks, threads, 0, stream>>>(uv, idcs, sid, tex0, tex1, out);
  }
}